// ChamferDistance_3813930959465
// MI455X (gfx1250) — compile-verified
//
#include <hip/hip_runtime.h>
#include <hip/hip_bf16.h>

// CDNA5 / gfx1250 Chamfer distance.
// d(i,j) = |q_i|^2 + |r_j|^2 - 2<q_i, r_j>
// A-row i = (-2x,-2y,-2z, 1), B-col j = (x',y',z', |r_j|^2)
//  => V_WMMA_F32_16X16X4_F32 produces M(i,j) = -2<q,r> + |r_j|^2 per 16x16 tile.
// Row-min over M, then add |q_i|^2 once at the end.

typedef __attribute__((ext_vector_type(2))) float v2f;
typedef __attribute__((ext_vector_type(8))) float v8f;

#define NPTS      8192
#define QPB       128          // queries per block (8 waves x 16 rows)
#define REF_CHUNK 2048         // refs staged in LDS per chunk (32 KB as float4)
#define NCHUNKS   (NPTS / REF_CHUNK)
#define NTILES    (REF_CHUNK / 16)

__global__ void chamfer_zero_kernel(float* out, int n) {
    int t = threadIdx.x;
    if (t < n) out[t] = 0.0f;
}

__global__ __launch_bounds__(256)
void chamfer_dir_kernel(const float* __restrict__ qry,
                        const float* __restrict__ ref,
                        float* __restrict__ out) {
    __shared__ float4 r4[REF_CHUNK];   // (x, y, z, |r|^2)
    __shared__ float4 q4[QPB];         // (-2x, -2y, -2z, 1)
    __shared__ float  n0s[QPB];        // |q|^2
    __shared__ float  wsum[8];

    const int tid   = threadIdx.x;
    const int b     = blockIdx.y;
    const int qbase = blockIdx.x * QPB;

    const int lane = tid & 31;
    const int w    = tid >> 5;       // wave id, 0..7
    const int half = lane >> 4;      // 0: K=0,1 / rows 0..7 side; 1: K=2,3 / rows 8..15 side
    const int m    = lane & 15;      // row (A) / column (B) index within tile

    // ---- stage this block's 128 query points ----
    if (tid < QPB) {
        const float* s = qry + ((size_t)b * NPTS + (qbase + tid)) * 3;
        float x = s[0], y = s[1], z = s[2];
        q4[tid]  = make_float4(-2.0f * x, -2.0f * y, -2.0f * z, 1.0f);
        n0s[tid] = x * x + y * y + z * z;
    }
    __syncthreads();

    // A-matrix registers for this wave's 16 query rows (16x4 f32 layout:
    // lanes 0-15 hold K=0,1; lanes 16-31 hold K=2,3 for row M=lane%16).
    float2 aq = *(reinterpret_cast<const float2*>(&q4[w * 16 + m]) + half);
    v2f Av; Av.x = aq.x; Av.y = aq.y;

    // running row-mins of M(i,j), in WMMA C/D layout
    float rm[8];
#pragma unroll
    for (int v = 0; v < 8; ++v) rm[v] = 3.4e38f;

    for (int c = 0; c < NCHUNKS; ++c) {
        __syncthreads();
        // ---- stage REF_CHUNK reference points into LDS with |r|^2 ----
        for (int p = tid; p < REF_CHUNK; p += 256) {
            const float* s = ref + ((size_t)b * NPTS + (c * REF_CHUNK + p)) * 3;
            float x = s[0], y = s[1], z = s[2];
            r4[p] = make_float4(x, y, z, x * x + y * y + z * z);
        }
        __syncthreads();

        // ---- sweep 16-wide ref tiles with f32 WMMA ----
#pragma unroll 4
        for (int t = 0; t < NTILES; ++t) {
            float2 bq = *(reinterpret_cast<const float2*>(&r4[t * 16 + m]) + half);
            v2f Bv; Bv.x = bq.x; Bv.y = bq.y;
            v8f cz = {};
            v8f d = __builtin_amdgcn_wmma_f32_16x16x4_f32(
                /*neg_a=*/false, Av, /*neg_b=*/false, Bv,
                /*c_mod=*/(short)0, cz, /*reuse_a=*/false, /*reuse_b=*/false);
#pragma unroll
            for (int v = 0; v < 8; ++v) rm[v] = fminf(rm[v], d[v]);
        }
    }

    // ---- finish: row-min across 16 lanes of each half, add |q|^2, sum rows ----
    float sum = 0.0f;
#pragma unroll
    for (int v = 0; v < 8; ++v) {
        float mv = rm[v];
        mv = fminf(mv, __shfl_xor(mv, 1, 32));
        mv = fminf(mv, __shfl_xor(mv, 2, 32));
        mv = fminf(mv, __shfl_xor(mv, 4, 32));
        mv = fminf(mv, __shfl_xor(mv, 8, 32));
        // half 0 lanes now hold min for row v; half 1 lanes for row v+8
        sum += mv + n0s[w * 16 + v + 8 * half];
    }
    // half0 lanes: rows 0..7 sum; half1 lanes: rows 8..15 sum -> combine
    sum += __shfl_xor(sum, 16, 32);

    if (lane == 0) wsum[w] = sum;
    __syncthreads();
    if (tid == 0) {
        float tot = 0.0f;
#pragma unroll
        for (int i = 0; i < 8; ++i) tot += wsum[i];
        atomicAdd(&out[b], tot * (1.0f / (float)NPTS));
    }
}

extern "C" void kernel_launch(void* const* d_in, const int* in_sizes, int n_in,
                              void* d_out, int out_size, void* d_ws, size_t ws_size,
                              hipStream_t stream) {
    const float* tmpl = (const float*)d_in[0];  // [4, 8192, 3] f32
    const float* srcp = (const float*)d_in[1];  // [4, 8192, 3] f32
    float* out = (float*)d_out;                 // [4] f32

    chamfer_zero_kernel<<<1, 32, 0, stream>>>(out, out_size);

    dim3 grid(NPTS / QPB, 4);
    // template -> source direction (mean over template of min over source)
    chamfer_dir_kernel<<<grid, 256, 0, stream>>>(tmpl, srcp, out);
    // source -> template direction
    chamfer_dir_kernel<<<grid, 256, 0, stream>>>(srcp, tmpl, out);
}